// dummyclass_11879879541471
// MI455X (gfx1250) — compile-verified
//
#include <hip/hip_runtime.h>
#include <hip/hip_bf16.h>

// Reference returns (input0, input1) unchanged -> d_out = input0 ++ input1.
// Pure streaming copy: 2 GiB HBM traffic, 0 FLOPs -> floor 2 GiB / 23.3 TB/s
// ~= 92 us. We use the CDNA5 async global<->LDS mover (ASYNCcnt-tracked,
// 512 B per wave32 instruction, no VGPR data) in a double-buffered pipeline.
// This revision strips all per-lane bounds checks (shapes divide exactly;
// uniform scalar branches only) and issues each tile's 8 async ops from one
// base address with constant offsets, so the compiler can use the shared
// 24-bit IOFFSET (added to BOTH the global and LDS address per the ISA)
// instead of rebuilding 64-bit addresses per op.

typedef float v4f __attribute__((ext_vector_type(4)));
typedef int   v4i __attribute__((ext_vector_type(4)));

#define BLOCK 256
#define KSLOT 8                           // float4 slots per thread per tile
#define TILE  ((long long)BLOCK * KSLOT)  // 2048 float4 = 32 KiB per tile

#if defined(__gfx1250__) &&                                             \
    __has_builtin(__builtin_amdgcn_global_load_async_to_lds_b128) &&    \
    __has_builtin(__builtin_amdgcn_global_store_async_from_lds_b128) && \
    __has_builtin(__builtin_amdgcn_s_wait_asynccnt)
#define USE_ASYNC_LDS 1
#else
#define USE_ASYNC_LDS 0
#endif

#if USE_ASYNC_LDS
#define GPTR(p) ((__attribute__((address_space(1))) v4i*)(p))
#define LPTR(p) ((__attribute__((address_space(3))) v4i*)(p))
#endif

// Main copy kernel: whole tiles only, no per-lane guards.
__global__ __launch_bounds__(BLOCK) void stream_copy_kernel(
    const v4f* __restrict__ src, v4f* __restrict__ dst, long long nTiles) {
#if USE_ASYNC_LDS
  __shared__ v4f lbuf[2][KSLOT * BLOCK];   // 2 x 32 KiB, per-thread slots
  const int tid = threadIdx.x;
  const long long g = gridDim.x;
  long long t = blockIdx.x;
  if (t >= nTiles) return;

  auto loadTile = [&](long long tt, int buf) {
    const v4f* ps = src + tt * TILE + tid;
#pragma unroll
    for (int k = 0; k < KSLOT; ++k)   // constant k*4096B offset, global==LDS
      __builtin_amdgcn_global_load_async_to_lds_b128(
          GPTR(ps + k * BLOCK), LPTR(&lbuf[buf][k * BLOCK + tid]), 0, 0);
  };
  auto storeTile = [&](long long tt, int buf) {
    v4f* pd = dst + tt * TILE + tid;
#pragma unroll
    for (int k = 0; k < KSLOT; ++k)
      __builtin_amdgcn_global_store_async_from_lds_b128(
          GPTR(pd + k * BLOCK), LPTR(&lbuf[buf][k * BLOCK + tid]), 0, 0);
  };

  // Ping-pong unrolled 2x so buffer indices are compile-time constants.
  loadTile(t, 0);
  for (;;) {
    __builtin_amdgcn_s_wait_asynccnt(0);   // buf0 loads done; buf1 stores done
    storeTile(t, 0);
    long long t1 = t + g;                  // uniform scalar branch only
    if (t1 >= nTiles) break;
    loadTile(t1, 1);                       // overlaps stores of buf0
    __builtin_amdgcn_s_wait_asynccnt(0);   // buf1 loads done; buf0 stores done
    storeTile(t1, 1);
    long long t2 = t1 + g;
    if (t2 >= nTiles) break;
    loadTile(t2, 0);                       // overlaps stores of buf1
    t = t2;
  }
  __builtin_amdgcn_s_wait_asynccnt(0);     // retire final stores
#else
  // Fallback: grid-stride non-temporal b128 copy over whole tiles.
  const long long n4 = nTiles * TILE;
  long long i = (long long)blockIdx.x * BLOCK + threadIdx.x;
  const long long stride = (long long)gridDim.x * BLOCK;
  for (; i < n4; i += stride) {
    v4f v = __builtin_nontemporal_load(src + i);
    __builtin_nontemporal_store(v, dst + i);
  }
#endif
}

// Generic guarded float copy (only used if shapes don't tile exactly;
// never launched for this harness's 2048x65536 shapes).
__global__ void scalar_copy_kernel(const float* __restrict__ src,
                                   float* __restrict__ dst, long long n) {
  long long i = (long long)blockIdx.x * blockDim.x + threadIdx.x;
  const long long stride = (long long)gridDim.x * blockDim.x;
  for (; i < n; i += stride) dst[i] = src[i];
}

extern "C" void kernel_launch(void* const* d_in, const int* in_sizes, int n_in,
                              void* d_out, int out_size, void* d_ws, size_t ws_size,
                              hipStream_t stream) {
  (void)n_in; (void)out_size; (void)d_ws; (void)ws_size;
  const long long nb = (long long)in_sizes[0];   // floats per input
  const long long tileF = TILE * 4;              // floats per tile (8192)

  if (nb % tileF == 0) {
    // Fast path (exact for 2048*65536): async-LDS tile pipeline.
    const v4f* in0 = (const v4f*)d_in[0];
    const v4f* in1 = (const v4f*)d_in[1];
    v4f* out = (v4f*)d_out;
    const long long n4 = nb / 4;
    const long long nTiles = n4 / TILE;          // 16384
    const int grid = (int)(nTiles < 4096 ? nTiles : 4096);
    stream_copy_kernel<<<grid, BLOCK, 0, stream>>>(in0, out, nTiles);
    stream_copy_kernel<<<grid, BLOCK, 0, stream>>>(in1, out + n4, nTiles);
  } else {
    // Robust path for non-tiling shapes.
    const float* in0 = (const float*)d_in[0];
    const float* in1 = (const float*)d_in[1];
    float* out = (float*)d_out;
    const int grid = 4096;
    scalar_copy_kernel<<<grid, BLOCK, 0, stream>>>(in0, out, nb);
    scalar_copy_kernel<<<grid, BLOCK, 0, stream>>>(in1, out + nb, nb);
  }
}